// RENBase_44444321579123
// MI455X (gfx1250) — compile-verified
//
#include <hip/hip_runtime.h>

// ---------------------------------------------------------------------------
// REN forward pass for MI455X (gfx1250), f32 WMMA (V_WMMA_F32_16X16X4_F32)
// + Tensor Data Mover staging for the equilibrium-layer diagonal blocks.
// NX=NV=256, NY=NU=128, BATCH=8192, n=2*NX+NV=768.
// ---------------------------------------------------------------------------

typedef __attribute__((ext_vector_type(2))) float v2f;
typedef __attribute__((ext_vector_type(8))) float v8f;

#define EPS_F32 1.1920928955078125e-07f

#if __has_builtin(__builtin_amdgcn_tensor_load_to_lds) && \
    __has_builtin(__builtin_amdgcn_s_wait_tensorcnt)
#define REN_HAS_TDM 1
typedef __attribute__((ext_vector_type(4))) unsigned int v4u;
typedef __attribute__((ext_vector_type(8))) int v8i;
typedef __attribute__((ext_vector_type(4))) int v4i;
#else
#define REN_HAS_TDM 0
#endif

// ---- WMMA helpers ----------------------------------------------------------
// acc += A(16xK, row-major, lda) * W(16xK, row-major, ldw)^T
// i.e. C[m,n] += sum_k A[m0+m, k] * W[n0+n, k]   (B-matrix of WMMA = W^T)
// A fragment: lane L: m = L&15, k-half = (L>>4)*2  -> {A[m,kh], A[m,kh+1]}
// B fragment: lane L: n = L&15, k-half = (L>>4)*2  -> {W[n,kh], W[n,kh+1]}
// lda/ldw are multiples of 4 and kh is even -> 8-byte aligned v2f (b64) loads.
__device__ __forceinline__ v8f wmma_acc_f32(const float* __restrict__ A, int lda,
                                            const float* __restrict__ W, int ldw,
                                            int K, v8f acc) {
  const int lane = threadIdx.x & 31;
  const int mn   = lane & 15;
  const int kh   = (lane >> 4) << 1;   // 0 or 2
  const float* ap = A + mn * lda + kh;
  const float* wp = W + mn * ldw + kh;
  for (int k = 0; k < K; k += 4) {
    v2f a = *(const v2f*)(ap + k);
    v2f b = *(const v2f*)(wp + k);
    acc = __builtin_amdgcn_wmma_f32_16x16x4_f32(false, a, false, b,
                                                (short)0, acc, false, false);
  }
  return acc;
}

// C/D layout: lane L holds column n=L&15, rows r + 8*(L>>4) in VGPR r.
__device__ __forceinline__ void store_tile_bias(float* __restrict__ C, int ldc,
                                                const float* __restrict__ bias,
                                                v8f acc) {
  const int lane = threadIdx.x & 31;
  const int n    = lane & 15;
  const int hi   = (lane >> 4) * 8;
  const float bn = bias[n];
#pragma unroll
  for (int r = 0; r < 8; ++r)
    C[(hi + r) * ldc + n] = acc[r] + bn;
}

// ---- 1) Frobenius norm of X (deterministic two-stage reduction) ------------
__global__ void sumsq_partial_kernel(const float* __restrict__ X,
                                     float* __restrict__ partials) {
  __shared__ float red[256];
  const int tid  = threadIdx.x;
  const int base = blockIdx.x * 2048;   // 288 blocks * 2048 = 768*768
  float s = 0.0f;
  for (int t = tid; t < 2048; t += 256) {
    float v = X[base + t];
    s += v * v;
  }
  red[tid] = s;
  __syncthreads();
  for (int off = 128; off > 0; off >>= 1) {
    if (tid < off) red[tid] += red[tid + off];
    __syncthreads();
  }
  if (tid == 0) partials[blockIdx.x] = red[0];
}

__global__ void scale_kernel(const float* __restrict__ partials,
                             const float* __restrict__ p,
                             float* __restrict__ scale) {
  if (threadIdx.x == 0) {
    float s = 0.0f;
    for (int i = 0; i < 288; ++i) s += partials[i];   // fixed order
    s = fmaxf(s, EPS_F32);                            // l2_norm(X)^2
    scale[0] = p[0] * p[0] / s;
  }
}

// ---- 2) H = scale * X^T X + eps*I  (768x768, WMMA) -------------------------
// grid (48, 12), block 128 (4 waves; wave w covers j-tile blockIdx.y*4+w)
__global__ void xtx_wmma_kernel(const float* __restrict__ X,
                                const float* __restrict__ scale,
                                float* __restrict__ H) {
  const int wave = threadIdx.x >> 5;
  const int i0   = blockIdx.x * 16;
  const int j0   = (blockIdx.y * 4 + wave) * 16;
  const int lane = threadIdx.x & 31;
  const int mn   = lane & 15;
  const int kh   = (lane >> 4) << 1;
  v8f acc = {};
  for (int k = 0; k < 768; k += 4) {
    // A = X^T : A[m,k] = X[k, i0+m];  B = X : B[k,n] = X[k, j0+n]
    v2f a = { X[(k + kh) * 768 + i0 + mn], X[(k + kh + 1) * 768 + i0 + mn] };
    v2f b = { X[(k + kh) * 768 + j0 + mn], X[(k + kh + 1) * 768 + j0 + mn] };
    acc = __builtin_amdgcn_wmma_f32_16x16x4_f32(false, a, false, b,
                                                (short)0, acc, false, false);
  }
  const float sc = scale[0];
  const int hi = (lane >> 4) * 8;
#pragma unroll
  for (int r = 0; r < 8; ++r) {
    const int gi = i0 + hi + r;
    const int gj = j0 + mn;
    H[gi * 768 + gj] = acc[r] * sc + ((gi == gj) ? EPS_F32 : 0.0f);
  }
}

// ---- 3) Build augmented system AUG = [E | H31 | H32 | B2]  (256 x 896) -----
__global__ void build_aug_kernel(const float* __restrict__ H,
                                 const float* __restrict__ Y1,
                                 const float* __restrict__ B2,
                                 float* __restrict__ aug) {
  const int idx = blockIdx.x * 256 + threadIdx.x;  // 896 blocks * 256 = 229376
  const int i = idx / 896;
  const int j = idx % 896;
  float v;
  if (j < 256) {
    // E = (H11 + H33 + Y1 - Y1^T) / 2
    v = 0.5f * (H[i * 768 + j] + H[(512 + i) * 768 + (512 + j)] +
                Y1[i * 256 + j] - Y1[j * 256 + i]);
  } else if (j < 512) {
    v = H[(512 + i) * 768 + (j - 256)];            // F = H31
  } else if (j < 768) {
    v = H[(512 + i) * 768 + 256 + (j - 512)];      // B1_imp = H32
  } else {
    v = B2[i * 128 + (j - 768)];                   // B2
  }
  aug[i * 896 + j] = v;
}

// ---- 4) C1_e, D11_e, D12_e (elementwise with Lambda_inv = 2/diag(H22)) -----
__global__ void build_explicit_kernel(const float* __restrict__ H,
                                      const float* __restrict__ D12,
                                      float* __restrict__ C1E,
                                      float* __restrict__ D11E,
                                      float* __restrict__ D12E) {
  const int idx = blockIdx.x * 256 + threadIdx.x;  // 640 blocks -> 163840
  if (idx < 65536) {
    const int i = idx >> 8, j = idx & 255;
    const float li = 2.0f / H[(256 + i) * 768 + 256 + i];
    C1E[idx] = -li * H[(256 + i) * 768 + j];                 // -Lambda_inv*H21
  } else if (idx < 131072) {
    const int t = idx - 65536;
    const int i = t >> 8, j = t & 255;
    const float li = 2.0f / H[(256 + i) * 768 + 256 + i];
    D11E[t] = (j < i) ? (-li * H[(256 + i) * 768 + 256 + j]) : 0.0f;
  } else {
    const int t = idx - 131072;                              // 32768 elems
    const int i = t >> 7;
    const float li = 2.0f / H[(256 + i) * 768 + 256 + i];
    D12E[t] = li * D12[t];
  }
}

// ---- 5) Gauss-Jordan solve of E * Z = RHS in place (single workgroup) ------
// aug: 256 x 896; after the loop, cols [256,896) hold [A_e | B1_e | B2_e].
__global__ void gauss_jordan_kernel(float* __restrict__ aug) {
  __shared__ float prow[896];
  __shared__ float fac[256];
  const int tid = threadIdx.x;  // 1024 threads
  for (int k = 0; k < 256; ++k) {
    __syncthreads();
    if (tid == 0) fac[0] = 1.0f / aug[k * 896 + k];
    __syncthreads();
    const float pinv = fac[0];
    for (int j = tid; j < 896; j += 1024) {
      const float v = aug[k * 896 + j] * pinv;
      aug[k * 896 + j] = v;
      prow[j] = v;
    }
    __syncthreads();
    if (tid < 256) fac[tid] = aug[tid * 896 + k];
    __syncthreads();
    const int r = tid >> 2;          // row 0..255
    const int c = tid & 3;           // column chunk of 224
    if (r != k) {
      const float f = fac[r];
      float* row = aug + r * 896;
      for (int j = c * 224; j < (c + 1) * 224; ++j)
        row[j] -= f * prow[j];
    }
  }
}

// ---- 6) b = state @ C1_e^T + inputs @ D12_e^T + bv  -> WB (8192 x 256) -----
// grid (512, 4), block 128 (4 waves over n-tiles)
__global__ void b_wmma_kernel(const float* __restrict__ state,
                              const float* __restrict__ inputs,
                              const float* __restrict__ C1E,
                              const float* __restrict__ D12E,
                              const float* __restrict__ bv,
                              float* __restrict__ WB) {
  const int wave = threadIdx.x >> 5;
  const int m0 = blockIdx.x * 16;
  const int n0 = (blockIdx.y * 4 + wave) * 16;
  v8f acc = {};
  acc = wmma_acc_f32(state  + m0 * 256, 256, C1E  + n0 * 256, 256, 256, acc);
  acc = wmma_acc_f32(inputs + m0 * 128, 128, D12E + n0 * 128, 128, 128, acc);
  store_tile_bias(WB + m0 * 256 + n0, 256, bv + n0, acc);
}

// ---- 7a) Equilibrium block GEMM: S = b_blk + w[:, <16I] @ D11_blk^T --------
// grid 128, block 128 (4 waves over m-tiles). Column block I (16 cols).
__global__ void s_wmma_kernel(const float* __restrict__ WB,
                              const float* __restrict__ D11E,
                              float* __restrict__ S, int I) {
  const int wave = threadIdx.x >> 5;
  const int m0 = (blockIdx.x * 4 + wave) * 16;
  const int K  = I * 16;
  v8f acc = {};
  acc = wmma_acc_f32(WB + m0 * 256, 256, D11E + K * 256, 256, K, acc);
  const int lane = threadIdx.x & 31;
  const int n  = lane & 15;
  const int hi = (lane >> 4) * 8;
#pragma unroll
  for (int r = 0; r < 8; ++r) {
    const int m = m0 + hi + r;
    S[m * 16 + n] = acc[r] + WB[m * 256 + K + n];   // + b block (still intact)
  }
}

// ---- 7b) In-block 16-step triangular relu recurrence; writes w over b ------
// grid 32, block 256: one thread per batch row. The 16x16 diagonal D11 block
// is staged into LDS by the Tensor Data Mover (2D tile of a 256x256 tensor),
// issued once per workgroup by wave 0; other waves sync on the barrier.
__global__ void relu_seq_kernel(const float* __restrict__ S,
                                const float* __restrict__ D11E,
                                float* __restrict__ WB, int I) {
  __shared__ float d[16][16];
  const int tid = threadIdx.x;
#if REN_HAS_TDM
  if (tid < 32) {
    // D# group 0: count=1 | lds_addr | 57-bit global_addr | type=2
    const unsigned long long ga =
        (unsigned long long)(uintptr_t)(D11E + (I * 16) * 256 + I * 16);
    const unsigned int lds_addr = (unsigned int)(uintptr_t)(&d[0][0]);
    v4u g0;
    g0[0] = 1u;                                          // count=1 (valid D#)
    g0[1] = lds_addr;                                    // lds_addr[31:0]
    g0[2] = (unsigned int)(ga & 0xffffffffu);            // global_addr[31:0]
    g0[3] = (unsigned int)((ga >> 32) & 0x01ffffffu)     // global_addr[56:32]
            | (2u << 30);                                // type=2 ("image")
    // D# group 1: data_size=4B; tensor 256x256, dim0_stride=256; tile 16x16
    v8i g1;
    g1[0] = (int)(2u << 16);        // data_size=2 (4 bytes), no mask/flags
    g1[1] = (int)(256u << 16);      // tensor_dim0[15:0]  in bits [31:16]
    g1[2] = (int)(256u << 16);      // tensor_dim0[31:16]=0 | tensor_dim1[15:0]
    g1[3] = (int)(16u << 16);       // tensor_dim1[31:16]=0 | tile_dim0=16
    g1[4] = 16;                     // tile_dim1=16, tile_dim2=0
    g1[5] = 256;                    // tensor_dim0_stride[31:0]
    g1[6] = 0;                      // stride hi bits / tensor_dim1_stride lo
    g1[7] = 0;
    const v4i z4 = {0, 0, 0, 0};            // groups 2/3 unused (2D tensor)
    const v8i z8 = {0, 0, 0, 0, 0, 0, 0, 0};
    __builtin_amdgcn_tensor_load_to_lds(g0, g1, z4, z4, z8, 0);
    __builtin_amdgcn_s_wait_tensorcnt(0);
  }
#else
  if (tid < 256)
    d[tid >> 4][tid & 15] =
        D11E[(I * 16 + (tid >> 4)) * 256 + I * 16 + (tid & 15)];
#endif
  __syncthreads();
  const int m = blockIdx.x * 256 + tid;
  float wv[16];
#pragma unroll
  for (int i = 0; i < 16; ++i) {
    float v = S[m * 16 + i];
#pragma unroll
    for (int j = 0; j < 16; ++j)
      if (j < i) v += wv[j] * d[i][j];
    wv[i] = fmaxf(v, 0.0f);
    WB[m * 256 + I * 16 + i] = wv[i];
  }
}

// ---- 8) x1 = state@A_e^T + w@B1_e^T + inputs@B2_e^T + bx  (8192 x 256) -----
// A_e/B1_e/B2_e live in AUG cols 256/512/768, ld = 896.
__global__ void x1_wmma_kernel(const float* __restrict__ state,
                               const float* __restrict__ WB,
                               const float* __restrict__ inputs,
                               const float* __restrict__ aug,
                               const float* __restrict__ bx,
                               float* __restrict__ x1) {
  const int wave = threadIdx.x >> 5;
  const int m0 = blockIdx.x * 16;
  const int n0 = (blockIdx.y * 4 + wave) * 16;
  v8f acc = {};
  acc = wmma_acc_f32(state  + m0 * 256, 256, aug + n0 * 896 + 256, 896, 256, acc);
  acc = wmma_acc_f32(WB     + m0 * 256, 256, aug + n0 * 896 + 512, 896, 256, acc);
  acc = wmma_acc_f32(inputs + m0 * 128, 128, aug + n0 * 896 + 768, 896, 128, acc);
  store_tile_bias(x1 + m0 * 256 + n0, 256, bx + n0, acc);
}

// ---- 9) y = state@C2^T + w@D21^T + inputs@D22^T + by  (8192 x 128) ---------
__global__ void y_wmma_kernel(const float* __restrict__ state,
                              const float* __restrict__ WB,
                              const float* __restrict__ inputs,
                              const float* __restrict__ C2,
                              const float* __restrict__ D21,
                              const float* __restrict__ D22,
                              const float* __restrict__ by,
                              float* __restrict__ y) {
  const int wave = threadIdx.x >> 5;
  const int m0 = blockIdx.x * 16;
  const int n0 = (blockIdx.y * 4 + wave) * 16;
  v8f acc = {};
  acc = wmma_acc_f32(state  + m0 * 256, 256, C2  + n0 * 256, 256, 256, acc);
  acc = wmma_acc_f32(WB     + m0 * 256, 256, D21 + n0 * 256, 256, 256, acc);
  acc = wmma_acc_f32(inputs + m0 * 128, 128, D22 + n0 * 128, 128, 128, acc);
  store_tile_bias(y + m0 * 128 + n0, 128, by + n0, acc);
}

// ---------------------------------------------------------------------------
extern "C" void kernel_launch(void* const* d_in, const int* in_sizes, int n_in,
                              void* d_out, int out_size, void* d_ws, size_t ws_size,
                              hipStream_t stream) {
  const float* state  = (const float*)d_in[0];   // 8192 x 256
  const float* inputs = (const float*)d_in[1];   // 8192 x 128
  const float* X      = (const float*)d_in[2];   // 768 x 768
  const float* p      = (const float*)d_in[3];   // 1
  const float* B2     = (const float*)d_in[4];   // 256 x 128
  const float* D12    = (const float*)d_in[5];   // 256 x 128
  const float* Y1     = (const float*)d_in[6];   // 256 x 256
  const float* C2     = (const float*)d_in[7];   // 128 x 256
  const float* D21    = (const float*)d_in[8];   // 128 x 256
  const float* D22    = (const float*)d_in[9];   // 128 x 128
  const float* bx     = (const float*)d_in[10];  // 256
  const float* bv     = (const float*)d_in[11];  // 256
  const float* by     = (const float*)d_in[12];  // 128

  float* out = (float*)d_out;                    // [x1 (8192*256) | y (8192*128)]

  // Workspace layout (floats); total ~3.21M floats (~12.9 MB).
  float* ws       = (float*)d_ws;
  float* partials = ws;                  // 288 (padded to 512)
  float* scale    = ws + 512;            // 1 (padded to 512)
  float* H        = ws + 1024;           // 768*768      = 589824
  float* AUG      = H + 589824;          // 256*896      = 229376
  float* C1E      = AUG + 229376;        // 256*256      = 65536
  float* D11E     = C1E + 65536;         // 256*256      = 65536
  float* D12E     = D11E + 65536;        // 256*128      = 32768
  float* S        = D12E + 32768;        // 8192*16      = 131072
  float* WB       = S + 131072;          // 8192*256     = 2097152 (b, then w in-place)

  // --- parameter transform ---
  sumsq_partial_kernel<<<288, 256, 0, stream>>>(X, partials);
  scale_kernel<<<1, 32, 0, stream>>>(partials, p, scale);
  xtx_wmma_kernel<<<dim3(48, 12), 128, 0, stream>>>(X, scale, H);
  build_aug_kernel<<<896, 256, 0, stream>>>(H, Y1, B2, AUG);
  build_explicit_kernel<<<640, 256, 0, stream>>>(H, D12, C1E, D11E, D12E);
  gauss_jordan_kernel<<<1, 1024, 0, stream>>>(AUG);

  // --- batch pre-activation: b -> WB ---
  b_wmma_kernel<<<dim3(512, 4), 128, 0, stream>>>(state, inputs, C1E, D12E, bv, WB);

  // --- blocked triangular equilibrium layer (w overwrites b in WB) ---
  for (int I = 0; I < 16; ++I) {
    s_wmma_kernel<<<128, 128, 0, stream>>>(WB, D11E, S, I);
    relu_seq_kernel<<<32, 256, 0, stream>>>(S, D11E, WB, I);
  }

  // --- outputs ---
  x1_wmma_kernel<<<dim3(512, 4), 128, 0, stream>>>(state, WB, inputs, AUG, bx, out);
  y_wmma_kernel<<<dim3(512, 2), 128, 0, stream>>>(state, WB, inputs, C2, D21, D22, by,
                                                  out + 8192 * 256);
  (void)in_sizes; (void)n_in; (void)out_size; (void)ws_size;
}